// MyModel_44444321579367
// MI455X (gfx1250) — compile-verified
//
#include <hip/hip_runtime.h>
#include <math.h>

// Problem constants (from reference)
#define B_     8
#define N_     4096
#define D_     64
#define R_     64
#define O_DIM  12
#define HID_   128
#define EPSF   1e-6f
#define TWO_PI 6.28318530717958647692f

// U-GEMM tiling
#define NSLICE 8            // N split across blocks (per batch)
#define CHUNK  64           // n-rows staged per double-buffer step
#define LROW   68           // padded LDS row stride in floats (272B, 16B-aligned)

typedef __attribute__((ext_vector_type(2))) float v2f;
typedef __attribute__((ext_vector_type(8))) float v8f;

__device__ __forceinline__ float softplusf(float x) {
    // logaddexp(x, 0) = max(x,0) + log1p(exp(-|x|))  (matches jax.nn.softplus)
    return fmaxf(x, 0.0f) + log1pf(expf(-fabsf(x)));
}
__device__ __forceinline__ float gelu_exact(float x) {
    return 0.5f * x * (1.0f + erff(x * 0.70710678118654752440f));
}

// ---------------------------------------------------------------------------
// Kernel 1: Bm = softplus(basis_raw)   (N x R)
// ---------------------------------------------------------------------------
__global__ void bm_kernel(const float* __restrict__ basis_raw, float* __restrict__ Bm) {
    int i = blockIdx.x * blockDim.x + threadIdx.x;
    if (i < N_ * R_) Bm[i] = softplusf(basis_raw[i]);
}

// ---------------------------------------------------------------------------
// Kernel 2: colsum[r] = sum_n Bm[n,r]  (deterministic LDS tree reduce)
// ---------------------------------------------------------------------------
__global__ void colsum_kernel(const float* __restrict__ Bm, float* __restrict__ colsum) {
    __shared__ float red[256];
    const int r = blockIdx.x, t = threadIdx.x;
    float acc = 0.0f;
    for (int n = t; n < N_; n += 256) acc += Bm[n * R_ + r];
    red[t] = acc;
    __syncthreads();
    for (int s = 128; s > 0; s >>= 1) {
        if (t < s) red[t] += red[t + s];
        __syncthreads();
    }
    if (t == 0) colsum[r] = red[0];
}

// ---------------------------------------------------------------------------
// Kernel 3: partial U[b] = Bm^T @ H[b] over an N-slice.
//   grid = B_*NSLICE blocks, 256 threads (8 waves).
//   Double-buffered async staging: GLOBAL_LOAD_ASYNC_TO_LDS_B128 pulls a
//   CHUNK x 64 tile of H and of Bm into padded LDS; s_wait_asynccnt + barrier;
//   8 waves = (r-tile 0..3) x (d-half 0..1) each run V_WMMA_F32_16X16X4_F32
//   over the staged chunk and accumulate privately -> direct store of partials.
// ---------------------------------------------------------------------------
__global__ void __launch_bounds__(256) u_gemm_kernel(const float* __restrict__ H,
                                                     const float* __restrict__ Bm,
                                                     float* __restrict__ Upart) {
    const int b     = blockIdx.x >> 3;
    const int slice = blockIdx.x & (NSLICE - 1);
    const int t     = threadIdx.x;
    const int lane  = t & 31;
    const int wave  = t >> 5;
    const int mr    = lane & 15;            // M for A, N(d) for B, per fp32 WMMA layout
    const int klo   = (lane < 16) ? 0 : 2;  // K pair owned by this half-wave
    const int rt    = wave & 3;             // r-tile
    const int dh    = wave >> 2;            // d half
    const int rb    = rt * 16;
    const int db    = dh * 32;

    __shared__ __attribute__((aligned(16))) float Hs[2][CHUNK * LROW];
    __shared__ __attribute__((aligned(16))) float Bs[2][CHUNK * LROW];

    const int n_base = slice * (N_ / NSLICE);
    const float* __restrict__ Hb = H + ((size_t)b * N_ + n_base) * D_;
    const float* __restrict__ Bb = Bm + (size_t)n_base * R_;

    // Stage one CHUNKx64 tile of H and Bm into LDS buffer `buf` (async).
    // 1024 x b128 per array; 4 per thread. Rows padded to LROW floats.
    auto stage = [&](int buf, int chunk) {
        const float* hsrc = Hb + (size_t)chunk * CHUNK * D_;
        const float* bsrc = Bb + (size_t)chunk * CHUNK * R_;
#pragma unroll
        for (int k = 0; k < 4; ++k) {
            const int idx = t + k * 256;
            const int row = idx >> 4;
            const int c4  = (idx & 15) << 2;
            const int loff = row * LROW + c4;
            const int goff = row * 64 + c4;
            unsigned hdst = (unsigned)(unsigned long long)(const void*)&Hs[buf][loff];
            const float* hg = hsrc + goff;
            asm volatile("global_load_async_to_lds_b128 %0, %1, off"
                         :: "v"(hdst), "v"(hg) : "memory");
            unsigned bdst = (unsigned)(unsigned long long)(const void*)&Bs[buf][loff];
            const float* bg = bsrc + goff;
            asm volatile("global_load_async_to_lds_b128 %0, %1, off"
                         :: "v"(bdst), "v"(bg) : "memory");
        }
    };

    v8f acc0 = {}, acc1 = {};

    stage(0, 0);
    const int NCH = (N_ / NSLICE) / CHUNK;  // 8 chunks per slice
    for (int c = 0; c < NCH; ++c) {
        const int cur = c & 1;
        asm volatile("s_wait_asynccnt 0x0" ::: "memory");  // my staged bytes are in LDS
        __syncthreads();                                   // everyone's bytes are in LDS
        if (c + 1 < NCH) stage(1 - cur, c + 1);            // overlap next stage w/ compute
        const float* __restrict__ hb = &Hs[cur][0];
        const float* __restrict__ bb = &Bs[cur][0];
#pragma unroll
        for (int ks = 0; ks < CHUNK / 4; ++ks) {
            const int r0 = (ks * 4 + klo) * LROW;
            v2f a;  a.x  = bb[r0 + rb + mr];       a.y  = bb[r0 + LROW + rb + mr];
            v2f b0; b0.x = hb[r0 + db + mr];       b0.y = hb[r0 + LROW + db + mr];
            v2f b1; b1.x = hb[r0 + db + 16 + mr];  b1.y = hb[r0 + LROW + db + 16 + mr];
            acc0 = __builtin_amdgcn_wmma_f32_16x16x4_f32(false, a, false, b0, (short)0, acc0, false, false);
            acc1 = __builtin_amdgcn_wmma_f32_16x16x4_f32(false, a, false, b1, (short)0, acc1, false, false);
        }
        __syncthreads();   // all waves finished reading buf[cur] before it is restaged
    }

    // Direct store of this block's 64x64 partial; each wave owns 16 rows x 32 cols.
    float* __restrict__ up = Upart + ((size_t)blockIdx.x << 12);  // 4096 floats/block
    const int mrow = (lane < 16) ? 0 : 8;
#pragma unroll
    for (int j = 0; j < 8; ++j) {
        const int row = (rb + j + mrow) << 6;
        up[row + db + mr]      = acc0[j];
        up[row + db + 16 + mr] = acc1[j];
    }
}

// ---------------------------------------------------------------------------
// Kernel 3b: U = sum of NSLICE partials (fixed order -> deterministic)
// ---------------------------------------------------------------------------
__global__ void u_reduce_kernel(const float* __restrict__ Upart, float* __restrict__ U) {
    const int i = blockIdx.x * blockDim.x + threadIdx.x;   // over B_*R_*D_ = 32768
    if (i >= B_ * R_ * D_) return;
    const int b = i >> 12;
    const int k = i & 4095;
    float s = 0.0f;
#pragma unroll
    for (int sl = 0; sl < NSLICE; ++sl)
        s += Upart[(((size_t)b * NSLICE + sl) << 12) + k];
    U[i] = s;
}

// ---------------------------------------------------------------------------
// Kernel 4: tiny head.  Per batch b:
//   v[r] = U[b,r,:]·w_out ; ctx[r] = sqrt(mean_d U^2 + eps)
//   per o: feat->gelu MLP->softplus s ; p'=c*s*v ; q=s*colsum ;
//          g = (1-c)*w_out + ctW·bt + cdW·bd ; off = b_out + bt·ctb + bd·cdb
// ---------------------------------------------------------------------------
__global__ void __launch_bounds__(128) head_kernel(
        const float* __restrict__ U, const float* __restrict__ colsum,
        const float* __restrict__ ts_out,
        const float* __restrict__ W1, const float* __restrict__ b1,
        const float* __restrict__ W2, const float* __restrict__ b2,
        const float* __restrict__ alpha_p, const float* __restrict__ fuse_p,
        const float* __restrict__ ctW, const float* __restrict__ ctb,
        const float* __restrict__ cdW, const float* __restrict__ cdb,
        const float* __restrict__ w_out, const float* __restrict__ b_out_p,
        float* __restrict__ pbuf, float* __restrict__ qbuf,
        float* __restrict__ gbuf, float* __restrict__ offbuf) {
    const int b = blockIdx.x;
    const int t = threadIdx.x;
    __shared__ float ctx_s[64], v_s[64], h_s[128], bt_s[16], bd_s[8];

    if (t < 64) {
        float ss = 0.0f, vd = 0.0f;
        const float* Ur = U + ((size_t)b * R_ + t) * D_;
        for (int d = 0; d < D_; ++d) { float u = Ur[d]; ss += u * u; vd += u * w_out[d]; }
        ctx_s[t] = sqrtf(ss * (1.0f / (float)D_) + EPSF);
        v_s[t]   = vd;
    }
    __syncthreads();

    const float a_cl = fminf(fmaxf(alpha_p[0], 0.0f), 1.0f);
    const float uf   = softplusf(fuse_p[0]);
    const float c    = (uf / (1.0f + uf)) * a_cl;

    for (int o = 0; o < O_DIM; ++o) {
        const float ts0 = ts_out[((size_t)b * O_DIM + o) * 2 + 0];
        const float ts1 = ts_out[((size_t)b * O_DIM + o) * 2 + 1];
        if (t < 16) {
            const int k = (t & 7) + 1;
            const float ang = TWO_PI * ts0 * (float)k;
            bt_s[t] = (t < 8) ? sinf(ang) : cosf(ang);
        } else if (t < 24) {
            const int tt = t - 16;
            const int k = (tt & 3) + 1;
            const float ang = TWO_PI * ts1 * (float)k;
            bd_s[tt] = (tt < 4) ? sinf(ang) : cosf(ang);
        }
        __syncthreads();
        {   // h = gelu(feat @ W1 + b1); feat = [ts0, ts1, ctx]
            float acc = b1[t] + ts0 * W1[t] + ts1 * W1[HID_ + t];
            for (int i = 0; i < 64; ++i) acc += ctx_s[i] * W1[(2 + i) * HID_ + t];
            h_s[t] = gelu_exact(acc);
        }
        __syncthreads();
        if (t < 64) {
            float acc = b2[t];
            for (int j = 0; j < HID_; ++j) acc += h_s[j] * W2[j * R_ + t];
            const float s = softplusf(acc) + EPSF;
            const size_t idx = ((size_t)b * O_DIM + o) * R_ + t;
            pbuf[idx] = c * s * v_s[t];
            qbuf[idx] = s * colsum[t];
        } else {
            const int d = t - 64;
            float g = (1.0f - c) * w_out[d];
            for (int p2 = 0; p2 < 16; ++p2) g += ctW[d * 16 + p2] * bt_s[p2];
            for (int p2 = 0; p2 < 8; ++p2)  g += cdW[d * 8 + p2] * bd_s[p2];
            gbuf[((size_t)b * O_DIM + o) * D_ + d] = g;
        }
        if (t == 0) {
            float off = b_out_p[0];
            for (int p2 = 0; p2 < 16; ++p2) off += bt_s[p2] * ctb[p2];
            for (int p2 = 0; p2 < 8; ++p2)  off += bd_s[p2] * cdb[p2];
            offbuf[(size_t)b * O_DIM + o] = off;
        }
        __syncthreads();
    }
}

// ---------------------------------------------------------------------------
// Kernel 5: y[b,o,n] = H[b,n]·g[b,o] + Bm[n]·p'[b,o]/(Bm[n]·q[b,o]+eps) + off
//   grid = B_*16 blocks x 256 threads; per-(b,o) vectors staged in LDS.
// ---------------------------------------------------------------------------
__global__ void __launch_bounds__(256) final_kernel(
        const float* __restrict__ H, const float* __restrict__ Bm,
        const float* __restrict__ pbuf, const float* __restrict__ qbuf,
        const float* __restrict__ gbuf, const float* __restrict__ offbuf,
        float* __restrict__ out) {
    const int b = blockIdx.x >> 4;
    const int n = (blockIdx.x & 15) * 256 + threadIdx.x;
    const int t = threadIdx.x;

    __shared__ float4 g4[O_DIM * 16], p4[O_DIM * 16], q4[O_DIM * 16];
    __shared__ float offs[O_DIM];
    {
        const float4* gg = (const float4*)(gbuf + (size_t)b * O_DIM * D_);
        const float4* pp = (const float4*)(pbuf + (size_t)b * O_DIM * R_);
        const float4* qq = (const float4*)(qbuf + (size_t)b * O_DIM * R_);
        for (int i = t; i < O_DIM * 16; i += 256) { g4[i] = gg[i]; p4[i] = pp[i]; q4[i] = qq[i]; }
        if (t < O_DIM) offs[t] = offbuf[(size_t)b * O_DIM + t];
    }
    __syncthreads();

    const float4* __restrict__ H4  = (const float4*)(H + ((size_t)b * N_ + n) * D_);
    const float4* __restrict__ Bm4 = (const float4*)(Bm + (size_t)n * R_);

    float aH[O_DIM], aP[O_DIM], aQ[O_DIM];
#pragma unroll
    for (int o = 0; o < O_DIM; ++o) { aH[o] = 0.0f; aP[o] = 0.0f; aQ[o] = 0.0f; }

    for (int i = 0; i < 16; ++i) {
        const float4 h = H4[i];
        const float4 m = Bm4[i];
#pragma unroll
        for (int o = 0; o < O_DIM; ++o) {
            const float4 g = g4[o * 16 + i];
            aH[o] += h.x * g.x + h.y * g.y + h.z * g.z + h.w * g.w;
            const float4 p = p4[o * 16 + i];
            aP[o] += m.x * p.x + m.y * p.y + m.z * p.z + m.w * p.w;
            const float4 q = q4[o * 16 + i];
            aQ[o] += m.x * q.x + m.y * q.y + m.z * q.z + m.w * q.w;
        }
    }
#pragma unroll
    for (int o = 0; o < O_DIM; ++o) {
        out[((size_t)b * O_DIM + o) * N_ + n] = aH[o] + aP[o] / (aQ[o] + EPSF) + offs[o];
    }
}

// ---------------------------------------------------------------------------
extern "C" void kernel_launch(void* const* d_in, const int* in_sizes, int n_in,
                              void* d_out, int out_size, void* d_ws, size_t ws_size,
                              hipStream_t stream) {
    (void)in_sizes; (void)n_in; (void)out_size; (void)ws_size;
    const float* H         = (const float*)d_in[0];
    const float* ts_out    = (const float*)d_in[1];
    const float* basis_raw = (const float*)d_in[2];
    const float* W1        = (const float*)d_in[3];
    const float* b1        = (const float*)d_in[4];
    const float* W2        = (const float*)d_in[5];
    const float* b2        = (const float*)d_in[6];
    const float* alpha     = (const float*)d_in[7];
    const float* fuse      = (const float*)d_in[8];
    const float* ctW       = (const float*)d_in[9];
    const float* ctb       = (const float*)d_in[10];
    const float* cdW       = (const float*)d_in[11];
    const float* cdb       = (const float*)d_in[12];
    const float* w_out     = (const float*)d_in[13];
    const float* b_out     = (const float*)d_in[14];
    float* out = (float*)d_out;

    // Workspace layout (floats); total ~2.3 MB.
    float* ws     = (float*)d_ws;
    float* Bm     = ws;                          // N*R              = 262144
    float* colsum = Bm + N_ * R_;                // 64
    float* Upart  = colsum + 64;                 // B*NSLICE*R*D     = 262144
    float* U      = Upart + B_ * NSLICE * R_ * D_; // B*R*D          = 32768
    float* pbuf   = U + B_ * R_ * D_;            // B*O*R            = 6144
    float* qbuf   = pbuf + B_ * O_DIM * R_;      // 6144
    float* gbuf   = qbuf + B_ * O_DIM * R_;      // 6144
    float* offbuf = gbuf + B_ * O_DIM * D_;      // 96

    bm_kernel<<<(N_ * R_ + 255) / 256, 256, 0, stream>>>(basis_raw, Bm);
    colsum_kernel<<<R_, 256, 0, stream>>>(Bm, colsum);
    u_gemm_kernel<<<B_ * NSLICE, 256, 0, stream>>>(H, Bm, Upart);
    u_reduce_kernel<<<(B_ * R_ * D_ + 255) / 256, 256, 0, stream>>>(Upart, U);
    head_kernel<<<B_, 128, 0, stream>>>(U, colsum, ts_out, W1, b1, W2, b2, alpha, fuse,
                                        ctW, ctb, cdW, cdb, w_out, b_out,
                                        pbuf, qbuf, gbuf, offbuf);
    final_kernel<<<B_ * 16, 256, 0, stream>>>(H, Bm, pbuf, qbuf, gbuf, offbuf, out);
}